// WassersteinLoss_51032801411670
// MI455X (gfx1250) — compile-verified
//
#include <hip/hip_runtime.h>
#include <hip/hip_bf16.h>
#include <math.h>

// ---------------------------------------------------------------------------
// WassersteinLoss on gfx1250:
//   pred = sigmoid(x1 - x0)  (2-channel softmax, channel 1)
//   sort pred[b,:] and target[b,:]  (value-bucket counting sort + LDS bitonic)
//   loss = mean_b mean_i (sorted_p - sorted_q)^2   (WMMA D = A*A^T diagonal)
// ---------------------------------------------------------------------------

typedef __attribute__((ext_vector_type(2))) float v2f;
typedef __attribute__((ext_vector_type(8))) float v8f;

#define NB    16
#define NPIX  (768 * 768)        // 589824 elements per sample
#define NARR  32                 // 16 pred arrays + 16 target arrays
#define KBINS 4096               // linear value buckets over [0,1]
#define CAP   1024               // max bucket size held in LDS (mean ~150)
#define TILE  4096               // elements per block for hist/scatter
#define TPB   256

// CDNA5 async global->LDS path (guarded; falls back to plain loads if absent)
#if defined(__has_builtin)
#if __has_builtin(__builtin_amdgcn_global_load_async_to_lds_b32) && \
    __has_builtin(__builtin_amdgcn_s_wait_asynccnt)
#define HAVE_ASYNC_LDS 1
#endif
#endif

// ---- value helpers (identical code path in hist + scatter => bitwise same) --
__device__ __forceinline__ float pred_val(const float* __restrict__ X, int b, int i) {
    float x0 = X[((size_t)b * 2 + 0) * NPIX + i];
    float x1 = X[((size_t)b * 2 + 1) * NPIX + i];
    float d  = x1 - x0;
    return 1.0f / (1.0f + expf(-d));   // == softmax(axis=1)[1] for C=2
}
__device__ __forceinline__ float get_val(const float* __restrict__ X,
                                         const float* __restrict__ T, int a, int i) {
    return (a < NB) ? pred_val(X, a, i) : T[(size_t)(a - NB) * NPIX + i];
}
__device__ __forceinline__ int bin_of(float v) {
    int b = (int)(v * (float)KBINS);
    return b < 0 ? 0 : (b > KBINS - 1 ? KBINS - 1 : b);
}

// ---- K0: zero bin counters -------------------------------------------------
__global__ void k_zero(unsigned* __restrict__ counts) {
    int i = blockIdx.x * TPB + threadIdx.x;
    if (i < NARR * KBINS) counts[i] = 0u;
}

// ---- K1: per-tile LDS histogram, flush with global atomics -----------------
__global__ void k_hist(const float* __restrict__ X, const float* __restrict__ T,
                       unsigned* __restrict__ counts) {
    __shared__ unsigned h[KBINS];
    const int a = blockIdx.y;
    for (int j = threadIdx.x; j < KBINS; j += TPB) h[j] = 0u;
    __syncthreads();
    const int base = blockIdx.x * TILE;
#pragma unroll
    for (int k = 0; k < TILE / TPB; ++k) {
        int i = base + k * TPB + threadIdx.x;
        atomicAdd(&h[bin_of(get_val(X, T, a, i))], 1u);
    }
    __syncthreads();
    for (int j = threadIdx.x; j < KBINS; j += TPB) {
        unsigned c = h[j];
        if (c) atomicAdd(&counts[a * KBINS + j], c);
    }
}

// ---- K2: per-array exclusive scan of 4096 bins (1024 thr, 4 bins each) -----
__global__ void k_scan(const unsigned* __restrict__ counts,
                       unsigned* __restrict__ offs, unsigned* __restrict__ curs) {
    __shared__ unsigned lds[1024];
    const int a = blockIdx.x, tid = threadIdx.x;
    unsigned c[4], s = 0;
#pragma unroll
    for (int j = 0; j < 4; ++j) { c[j] = counts[a * KBINS + tid * 4 + j]; s += c[j]; }
    lds[tid] = s;
    __syncthreads();
    for (int off = 1; off < 1024; off <<= 1) {       // Hillis-Steele inclusive
        unsigned v = (tid >= off) ? lds[tid - off] : 0u;
        __syncthreads();
        lds[tid] += v;
        __syncthreads();
    }
    unsigned run = lds[tid] - s;                     // exclusive for this thread
#pragma unroll
    for (int j = 0; j < 4; ++j) {
        offs[a * KBINS + tid * 4 + j] = run;
        curs[a * KBINS + tid * 4 + j] = run;
        run += c[j];
    }
}

// ---- K3: scatter values into their bucket regions --------------------------
__global__ void k_scatter(const float* __restrict__ X, const float* __restrict__ T,
                          unsigned* __restrict__ curs, float* __restrict__ vals) {
    const int a = blockIdx.y;
    const int base = blockIdx.x * TILE;
#pragma unroll
    for (int k = 0; k < TILE / TPB; ++k) {
        int i = base + k * TPB + threadIdx.x;
        float v = get_val(X, T, a, i);
        unsigned pos = atomicAdd(&curs[a * KBINS + bin_of(v)], 1u);
        vals[(size_t)a * NPIX + pos] = v;
    }
}

// ---- K4: bitonic sort of each bucket in LDS (pow2-padded with +INF) --------
__global__ void k_sort(const unsigned* __restrict__ counts,
                       const unsigned* __restrict__ offs, float* __restrict__ vals) {
    __shared__ float buf[CAP];
    const int a = blockIdx.y, bin = blockIdx.x;
    unsigned cnt = counts[a * KBINS + bin];
    if (cnt <= 1) return;
    if (cnt > CAP) cnt = CAP;                        // statistically unreachable
    const unsigned start = offs[a * KBINS + bin];
    float* src = vals + (size_t)a * NPIX + start;
    unsigned M = 1; while (M < cnt) M <<= 1;         // uniform across block

#if HAVE_ASYNC_LDS
    // CDNA5: DMA bucket elements straight into LDS (no VGPR round-trip),
    // pad the pow2 tail with +INF via normal stores.
    for (unsigned i = threadIdx.x; i < M; i += TPB) {
        if (i < cnt) {
            __builtin_amdgcn_global_load_async_to_lds_b32(
                (__attribute__((address_space(1))) int*)(src + i),
                (__attribute__((address_space(3))) int*)&buf[i], 0, 0);
        } else {
            buf[i] = __int_as_float(0x7f800000);
        }
    }
    __builtin_amdgcn_s_wait_asynccnt(0);   // this wave's async writes done
    __syncthreads();                       // all waves' writes visible
#else
    for (unsigned i = threadIdx.x; i < M; i += TPB)
        buf[i] = (i < cnt) ? src[i] : __int_as_float(0x7f800000);
    __syncthreads();
#endif

    for (unsigned k = 2; k <= M; k <<= 1)
        for (unsigned j = k >> 1; j > 0; j >>= 1) {
            for (unsigned i = threadIdx.x; i < M; i += TPB) {
                unsigned ixj = i ^ j;
                if (ixj > i) {
                    float x = buf[i], y = buf[ixj];
                    bool up = ((i & k) == 0);
                    if (up ? (x > y) : (x < y)) { buf[i] = y; buf[ixj] = x; }
                }
            }
            __syncthreads();
        }
    for (unsigned i = threadIdx.x; i < cnt; i += TPB) src[i] = buf[i];
}

// ---- K5: WMMA reduction of matched squared differences ---------------------
// Same VGPR pair as SRC0 (A 16x4) and SRC1 (B 4x16): identical lane/VGPR
// mapping means B == A^T, so D = A*A^T; diagonal accumulates sum of squares.
#define TILES_PER_SAMPLE (NPIX / 64)     // 9216
#define TILES_PER_WAVE   36              // 36 | 9216 -> wave never crosses sample
#define TOTAL_WAVES      ((NB * TILES_PER_SAMPLE) / TILES_PER_WAVE)   // 4096
#define K5_BLOCKS        (TOTAL_WAVES / (TPB / 32))                   // 512

__global__ void k_wmma(const float* __restrict__ vals, float* __restrict__ parts) {
    const int wave = blockIdx.x * (TPB / 32) + (threadIdx.x >> 5);
    const int lane = threadIdx.x & 31;
    const int t0   = wave * TILES_PER_WAVE;
    const int b    = t0 / TILES_PER_SAMPLE;
    const float* __restrict__ P = vals + (size_t)b * NPIX;            // sorted pred
    const float* __restrict__ Q = vals + (size_t)(NB + b) * NPIX;     // sorted target
    const int local0 = (t0 - b * TILES_PER_SAMPLE) * 64;
    // A-matrix 16x4 f32 layout: lane l, vgpr j -> element (m = l&15, k = 2*(l>>4)+j)
    const int lidx = (lane & 15) * 4 + (lane >> 4) * 2;
    v8f c = {0.f, 0.f, 0.f, 0.f, 0.f, 0.f, 0.f, 0.f};
#pragma unroll 4
    for (int t = 0; t < TILES_PER_WAVE; ++t) {
        int off = local0 + t * 64 + lidx;
        v2f p = *(const v2f*)(P + off);
        v2f q = *(const v2f*)(Q + off);
        v2f a; a.x = p.x - q.x; a.y = p.y - q.y;
        c = __builtin_amdgcn_wmma_f32_16x16x4_f32(false, a, false, a,
                                                  (short)0, c, false, false);
    }
    // Diagonal of D: VGPR j holds (j,j) at lane j and (j+8,j+8) at lane j+24.
    float s = 0.0f;
#pragma unroll
    for (int j = 0; j < 8; ++j)
        if (lane == j || lane == j + 24) s = c[j];
    for (int m = 16; m; m >>= 1) s += __shfl_xor(s, m, 32);
    if (lane == 0) parts[wave] = s;
}

// ---- K6: deterministic final tree reduction --------------------------------
__global__ void k_final(const float* __restrict__ parts, float* __restrict__ out) {
    __shared__ float lds[1024];
    const int tid = threadIdx.x;
    float s = parts[tid] + parts[tid + 1024] + parts[tid + 2048] + parts[tid + 3072];
    lds[tid] = s;
    __syncthreads();
    for (int off = 512; off; off >>= 1) {
        if (tid < off) lds[tid] += lds[tid + off];
        __syncthreads();
    }
    if (tid == 0) out[0] = lds[0] * (1.0f / ((float)NB * (float)NPIX));
}

// ---------------------------------------------------------------------------
extern "C" void kernel_launch(void* const* d_in, const int* in_sizes, int n_in,
                              void* d_out, int out_size, void* d_ws, size_t ws_size,
                              hipStream_t stream) {
    const float* X = (const float*)d_in[0];   // [16, 2, 768, 768] logits
    const float* T = (const float*)d_in[1];   // [16, 768, 768] target
    float* out = (float*)d_out;               // scalar loss

    // Workspace layout (~77.1 MB total):
    float*    vals   = (float*)d_ws;                              // 32*NPIX f32
    unsigned* counts = (unsigned*)(vals + (size_t)NARR * NPIX);   // 32*4096 u32
    unsigned* offs   = counts + NARR * KBINS;                     // 32*4096 u32
    unsigned* curs   = offs   + NARR * KBINS;                     // 32*4096 u32
    float*    parts  = (float*)(curs + NARR * KBINS);             // 4096 f32

    k_zero   <<<(NARR * KBINS + TPB - 1) / TPB, TPB, 0, stream>>>(counts);
    k_hist   <<<dim3(NPIX / TILE, NARR), TPB, 0, stream>>>(X, T, counts);
    k_scan   <<<NARR, 1024, 0, stream>>>(counts, offs, curs);
    k_scatter<<<dim3(NPIX / TILE, NARR), TPB, 0, stream>>>(X, T, curs, vals);
    k_sort   <<<dim3(KBINS, NARR), TPB, 0, stream>>>(counts, offs, vals);
    k_wmma   <<<K5_BLOCKS, TPB, 0, stream>>>(vals, parts);
    k_final  <<<1, 1024, 0, stream>>>(parts, out);
}